// HeterogeneousGNN_37563783971457
// MI455X (gfx1250) — compile-verified
//
#include <hip/hip_runtime.h>

// HANConv forward for MI455X (gfx1250, wave32).
// All 128-wide GEMMs run on V_WMMA_F32_16X16X4_F32 (fp32 WMMA, full precision).
// Edge softmax/aggregation: 3-pass COO with f32 atomics (L2-resident tables).

typedef __attribute__((ext_vector_type(2))) float v2f;
typedef __attribute__((ext_vector_type(4))) float v4f;
typedef __attribute__((ext_vector_type(8))) float v8f;

#define HIDC  128
#define NHEAD 4
#define HD    32
#define NEDGE 400000
#define SLOPE 0.2f

#define ACT_NONE    0
#define ACT_RELU    1
#define ACT_TANH_CS 2

// monotone float <-> uint mapping so unsigned atomicMax == float max
__device__ __forceinline__ unsigned enc_f(float f) {
  unsigned u = __float_as_uint(f);
  return (u & 0x80000000u) ? ~u : (u | 0x80000000u);
}
__device__ __forceinline__ float dec_f(unsigned e) {
  return (e & 0x80000000u) ? __uint_as_float(e ^ 0x80000000u) : __uint_as_float(~e);
}

__global__ void fill_zero_k(float* __restrict__ p, long n) {
  long i = (long)blockIdx.x * blockDim.x + threadIdx.x;
  if (i < n) p[i] = 0.0f;
}

// C[M,128] = act(A[M,128] @ B[128,128] + bias)
// act==ACT_TANH_CS: atomicAdd per-column sums of tanh(.) into csum[128] (no C store).
__global__ void __launch_bounds__(256) gemm128_wmma_k(
    const float* __restrict__ A, const float* __restrict__ B,
    const float* __restrict__ bias, float* __restrict__ C,
    float* __restrict__ csum, int M, int act) {
  __shared__ float As[16 * 132];           // 16x128 A tile, padded stride
  const int tid  = threadIdx.x;
  const int wave = tid >> 5;               // 0..7 -> 16-col slice
  const int lane = tid & 31;
  const int lg   = lane >> 4;              // lane group 0/1
  const int lm   = lane & 15;
  const int row0 = blockIdx.x << 4;

  for (int i = tid; i < 16 * 32; i += 256) {          // 512 x float4
    int r = i >> 5;
    int c = (i & 31) << 2;
    int gr = row0 + r; if (gr >= M) gr = M - 1;       // clamp (tail rows unused)
    *(v4f*)&As[r * 132 + c] = *(const v4f*)&A[(long)gr * HIDC + c];
  }
  __syncthreads();

  const int n = (wave << 4) + lm;                     // output column
  v8f acc = {};
#pragma unroll 4
  for (int k0 = 0; k0 < HIDC; k0 += 4) {
    const int ka = k0 + (lg << 1);                    // K split across lane groups
    v2f a; a.x = As[lm * 132 + ka];      a.y = As[lm * 132 + ka + 1];
    v2f b; b.x = B[ka * HIDC + n];       b.y = B[(ka + 1) * HIDC + n];
    acc = __builtin_amdgcn_wmma_f32_16x16x4_f32(false, a, false, b,
                                                (short)0, acc, false, false);
  }
  const float bn = bias ? bias[n] : 0.0f;
  if (act == ACT_TANH_CS) {
    float s = 0.0f;
#pragma unroll
    for (int v = 0; v < 8; ++v) {
      int m = row0 + v + (lg << 3);
      if (m < M) s += tanhf(acc[v] + bn);
    }
    atomicAdd(&csum[n], s);
  } else {
#pragma unroll
    for (int v = 0; v < 8; ++v) {
      int m = row0 + v + (lg << 3);
      if (m < M) {
        float x = acc[v] + bn;
        if (act == ACT_RELU) x = fmaxf(x, 0.0f);
        C[(long)m * HIDC + n] = x;
      }
    }
  }
}

// out[n,h] = dot(h[n, h*32 : h*32+32], att[h*32 : ...])
__global__ void att_score_k(const float* __restrict__ hx, const float* __restrict__ att,
                            float* __restrict__ out, int N) {
  int t = blockIdx.x * blockDim.x + threadIdx.x;
  if (t >= N * NHEAD) return;
  int n = t >> 2, hd = t & 3;
  const v4f* hp = (const v4f*)&hx[(long)n * HIDC + hd * HD];
  const v4f* ap = (const v4f*)&att[hd * HD];
  float s = 0.0f;
#pragma unroll
  for (int i = 0; i < 8; ++i) {
    v4f a = hp[i], b = ap[i];
    s += a.x * b.x + a.y * b.y + a.z * b.z + a.w * b.w;
  }
  out[t] = s;
}

__global__ void edge_alpha_k(const int* __restrict__ si, const int* __restrict__ di,
                             const float* __restrict__ as_, const float* __restrict__ ad_,
                             float* __restrict__ alpha, unsigned* __restrict__ menc, int E) {
  int e = blockIdx.x * blockDim.x + threadIdx.x;
  if (e >= E) return;
  int s = si[e], d = di[e];
  v4f a = *(const v4f*)&as_[s * 4];
  v4f b = *(const v4f*)&ad_[d * 4];
  v4f v = a + b;
#pragma unroll
  for (int hh = 0; hh < 4; ++hh) {
    float x = v[hh];
    x = (x > 0.0f) ? x : SLOPE * x;          // leaky relu
    alpha[e * 4 + hh] = x;
    atomicMax(&menc[d * 4 + hh], enc_f(x));  // segment max
  }
}

__global__ void edge_expsum_k(const int* __restrict__ di, const unsigned* __restrict__ menc,
                              float* __restrict__ alpha, float* __restrict__ den, int E) {
  int e = blockIdx.x * blockDim.x + threadIdx.x;
  if (e >= E) return;
  int d = di[e];
#pragma unroll
  for (int hh = 0; hh < 4; ++hh) {
    float m  = dec_f(menc[d * 4 + hh]);
    float ex = __expf(alpha[e * 4 + hh] - m);
    alpha[e * 4 + hh] = ex;
    atomicAdd(&den[d * 4 + hh], ex);
  }
}

// one thread per (edge, 16B chunk of the 512B message)
__global__ void edge_scatter_k(const int* __restrict__ si, const int* __restrict__ di,
                               const float* __restrict__ hsrc, const float* __restrict__ exv,
                               const float* __restrict__ den, float* __restrict__ agg, int E) {
  long t = (long)blockIdx.x * blockDim.x + threadIdx.x;
  if (t >= (long)E * 32) return;
  int e   = (int)(t >> 5);
  int idx = (int)(t & 31);
  int hh  = idx >> 3;
  int s = si[e], d = di[e];
  float w = exv[e * 4 + hh] / (den[d * 4 + hh] + 1e-16f);
  v4f hv = *(const v4f*)&hsrc[(long)s * HIDC + idx * 4];
  float* dst = &agg[(long)d * HIDC + idx * 4];
  atomicAdd(dst + 0, hv.x * w);
  atomicAdd(dst + 1, hv.y * w);
  atomicAdd(dst + 2, hv.z * w);
  atomicAdd(dst + 3, hv.w * w);
}

__global__ void relu_k(float* __restrict__ p, long n4) {
  long i = (long)blockIdx.x * blockDim.x + threadIdx.x;
  if (i >= n4) return;
  v4f v = *(v4f*)&p[i * 4];
  v.x = fmaxf(v.x, 0.0f); v.y = fmaxf(v.y, 0.0f);
  v.z = fmaxf(v.z, 0.0f); v.w = fmaxf(v.w, 0.0f);
  *(v4f*)&p[i * 4] = v;
}

// semantic attention scores + per-type softmax -> beta[7]
__global__ void beta_k(const float* __restrict__ ksum, const float* __restrict__ q,
                       float* __restrict__ beta) {
  __shared__ float red[128];
  __shared__ float score[8];
  const float invN[7] = {1.0f/100000.0f, 1.0f/100000.0f, 1.0f/100000.0f, 1.0f/100000.0f,
                         1.0f/50000.0f, 1.0f/5000.0f, 1.0f/2000.0f};
  int f = threadIdx.x;
  float qf = q[f];
  for (int r = 0; r < 7; ++r) {
    red[f] = ksum[r * 128 + f] * qf;
    __syncthreads();
    for (int st = 64; st > 0; st >>= 1) {
      if (f < st) red[f] += red[f + st];
      __syncthreads();
    }
    if (f == 0) score[r] = red[0] * invN[r];
    __syncthreads();
  }
  if (f == 0) {
    float mx = fmaxf(fmaxf(score[0], score[1]), fmaxf(score[2], score[3]));
    float b[4], s = 0.0f;
    for (int i = 0; i < 4; ++i) { b[i] = __expf(score[i] - mx); s += b[i]; }
    for (int i = 0; i < 4; ++i) beta[i] = b[i] / s;
    beta[4] = 1.0f; beta[5] = 1.0f; beta[6] = 1.0f;   // singleton softmax = 1
  }
}

// out = relu(sum_j beta[s0+j] * z_j)
__global__ void fused_k(const float* __restrict__ z0, const float* __restrict__ z1,
                        const float* __restrict__ z2, const float* __restrict__ z3,
                        const float* __restrict__ beta, int s0, int nslots,
                        float* __restrict__ out, long n4) {
  long i = (long)blockIdx.x * blockDim.x + threadIdx.x;
  if (i >= n4) return;
  v4f acc = beta[s0] * *(const v4f*)&z0[i * 4];
  if (nslots > 1) acc += beta[s0 + 1] * *(const v4f*)&z1[i * 4];
  if (nslots > 2) acc += beta[s0 + 2] * *(const v4f*)&z2[i * 4];
  if (nslots > 3) acc += beta[s0 + 3] * *(const v4f*)&z3[i * 4];
  acc.x = fmaxf(acc.x, 0.0f); acc.y = fmaxf(acc.y, 0.0f);
  acc.z = fmaxf(acc.z, 0.0f); acc.w = fmaxf(acc.w, 0.0f);
  *(v4f*)&out[i * 4] = acc;
}

extern "C" void kernel_launch(void* const* d_in, const int* in_sizes, int n_in,
                              void* d_out, int out_size, void* d_ws, size_t ws_size,
                              hipStream_t stream) {
  (void)in_sizes; (void)n_in; (void)out_size; (void)ws_size;
  const float* x_in[4] = {(const float*)d_in[0], (const float*)d_in[1],
                          (const float*)d_in[2], (const float*)d_in[3]};
  const int*   esrc    = (const int*)d_in[4];
  const int*   edst    = (const int*)d_in[5];
  const float* lin_w   = (const float*)d_in[6];
  const float* lin_b   = (const float*)d_in[7];
  const float* attS    = (const float*)d_in[8];
  const float* attD    = (const float*)d_in[9];
  const float* klin_w  = (const float*)d_in[10];
  const float* klin_b  = (const float*)d_in[11];
  const float* qv      = (const float*)d_in[12];
  const float* out_w   = (const float*)d_in[13];
  const float* out_b   = (const float*)d_in[14];
  float* out = (float*)d_out;
  float* ws  = (float*)d_ws;

  const int NN[4] = {100000, 50000, 5000, 2000};
  const int RS[7] = {0, 1, 2, 3, 0, 0, 0};
  const int RD[7] = {0, 0, 0, 0, 1, 2, 3};

  // workspace layout (float offsets)
  size_t off = 0;
  size_t XO[4], HO[4], ZO[7], ASO[7], ADO[7], MO[7], DEO[7], ALO[7];
  for (int t = 0; t < 4; ++t) { XO[t] = off; off += (size_t)NN[t] * HIDC; }
  for (int t = 0; t < 4; ++t) { HO[t] = off; off += (size_t)NN[t] * HIDC; }
  for (int r = 0; r < 7; ++r) { ZO[r]  = off; off += (size_t)NN[RD[r]] * HIDC; }
  for (int r = 0; r < 7; ++r) { ASO[r] = off; off += (size_t)NN[RS[r]] * 4; }
  for (int r = 0; r < 7; ++r) { ADO[r] = off; off += (size_t)NN[RD[r]] * 4; }
  for (int r = 0; r < 7; ++r) { MO[r]  = off; off += (size_t)NN[RD[r]] * 4; }
  for (int r = 0; r < 7; ++r) { DEO[r] = off; off += (size_t)NN[RD[r]] * 4; }
  for (int r = 0; r < 7; ++r) { ALO[r] = off; off += (size_t)NEDGE * 4; }
  size_t KSO = off; off += 7 * HIDC;
  size_t BO  = off; off += 8;

  auto cdiv = [](long a, long b) { return (int)((a + b - 1) / b); };

  for (int l = 0; l < 2; ++l) {
    fill_zero_k<<<cdiv(7 * HIDC, 256), 256, 0, stream>>>(ws + KSO, 7 * HIDC);

    // per-type projections: h_t = x_t @ lin_w[l,t] + lin_b[l,t]
    for (int t = 0; t < 4; ++t) {
      const float* xa = (l == 0) ? x_in[t] : ws + XO[t];
      gemm128_wmma_k<<<cdiv(NN[t], 16), 256, 0, stream>>>(
          xa, lin_w + (size_t)(l * 4 + t) * HIDC * HIDC,
          lin_b + (size_t)(l * 4 + t) * HIDC, ws + HO[t], nullptr, NN[t], ACT_NONE);
    }

    // per-relation edge attention + aggregation
    for (int r = 0; r < 7; ++r) {
      int ns = NN[RS[r]], nd = NN[RD[r]];
      att_score_k<<<cdiv((long)ns * 4, 256), 256, 0, stream>>>(
          ws + HO[RS[r]], attS + (size_t)(l * 7 + r) * HIDC, ws + ASO[r], ns);
      att_score_k<<<cdiv((long)nd * 4, 256), 256, 0, stream>>>(
          ws + HO[RD[r]], attD + (size_t)(l * 7 + r) * HIDC, ws + ADO[r], nd);
      fill_zero_k<<<cdiv((long)nd * 4, 256), 256, 0, stream>>>(ws + MO[r],  (long)nd * 4);
      fill_zero_k<<<cdiv((long)nd * 4, 256), 256, 0, stream>>>(ws + DEO[r], (long)nd * 4);
      fill_zero_k<<<cdiv((long)nd * HIDC, 256), 256, 0, stream>>>(ws + ZO[r], (long)nd * HIDC);
      edge_alpha_k<<<cdiv(NEDGE, 256), 256, 0, stream>>>(
          esrc + (size_t)r * NEDGE, edst + (size_t)r * NEDGE,
          ws + ASO[r], ws + ADO[r], ws + ALO[r], (unsigned*)(ws + MO[r]), NEDGE);
      edge_expsum_k<<<cdiv(NEDGE, 256), 256, 0, stream>>>(
          edst + (size_t)r * NEDGE, (const unsigned*)(ws + MO[r]),
          ws + ALO[r], ws + DEO[r], NEDGE);
      edge_scatter_k<<<cdiv((long)NEDGE * 32, 256), 256, 0, stream>>>(
          esrc + (size_t)r * NEDGE, edst + (size_t)r * NEDGE,
          ws + HO[RS[r]], ws + ALO[r], ws + DEO[r], ws + ZO[r], NEDGE);
    }

    // per-relation ReLU over the whole contiguous z region
    long ztot = 0; for (int r = 0; r < 7; ++r) ztot += (long)NN[RD[r]] * HIDC;
    relu_k<<<cdiv(ztot / 4, 256), 256, 0, stream>>>(ws + ZO[0], ztot / 4);

    // semantic attention: fused tanh(z @ klin_w + b) column sums
    for (int r = 0; r < 7; ++r) {
      gemm128_wmma_k<<<cdiv(NN[RD[r]], 16), 256, 0, stream>>>(
          ws + ZO[r], klin_w + (size_t)l * HIDC * HIDC,
          klin_b + (size_t)l * HIDC, nullptr, ws + KSO + (size_t)r * HIDC,
          NN[RD[r]], ACT_TANH_CS);
    }
    beta_k<<<1, 128, 0, stream>>>(ws + KSO, qv + (size_t)l * HIDC, ws + BO);

    // fuse relations per destination type -> new x
    fused_k<<<cdiv((long)NN[0] * HIDC / 4, 256), 256, 0, stream>>>(
        ws + ZO[0], ws + ZO[1], ws + ZO[2], ws + ZO[3],
        ws + BO, 0, 4, ws + XO[0], (long)NN[0] * HIDC / 4);
    for (int t = 1; t < 4; ++t) {
      int r = 3 + t;  // slots 4,5,6
      fused_k<<<cdiv((long)NN[t] * HIDC / 4, 256), 256, 0, stream>>>(
          ws + ZO[r], ws + ZO[r], ws + ZO[r], ws + ZO[r],
          ws + BO, r, 1, ws + XO[t], (long)NN[t] * HIDC / 4);
    }
  }

  // output heads, concatenated in type order
  size_t oo = 0;
  for (int t = 0; t < 4; ++t) {
    gemm128_wmma_k<<<cdiv(NN[t], 16), 256, 0, stream>>>(
        ws + XO[t], out_w + (size_t)t * HIDC * HIDC,
        out_b + (size_t)t * HIDC, out + oo, nullptr, NN[t], ACT_NONE);
    oo += (size_t)NN[t] * HIDC;
  }
}